// Temporal_Attention_layer_64965675319970
// MI455X (gfx1250) — compile-verified
//
#include <hip/hip_runtime.h>
#include <hip/hip_bf16.h>

#define BB 16
#define NN 512
#define FF 64
#define TT 512

// Column-major LDS stride for the 512x32 S/E strip: pad by 2 so it is
// (a) bank-conflict-free for the staging writes / softmax column walks
//     (bank = (2*col + row) % 64, distinct across 32 lanes), and
// (b) even, so v2f B-fragments are 8-byte aligned -> single ds_load_b64.
#define LSTR 514

typedef __attribute__((ext_vector_type(2))) float v2f;
typedef __attribute__((ext_vector_type(4))) float v4f;
typedef __attribute__((ext_vector_type(8))) float v8f;

// ---------------------------------------------------------------------------
// Kernel 1: S[b,n,t] = sigmoid( sum_f W3[f]*x[b,n,f,t] + bias[n,t] )
// Streams x (1.07 GB) exactly once -> non-temporal loads keep L2 free for
// S and Vs which kernel 2 re-reads. float4 vectorized, coalesced.
// ---------------------------------------------------------------------------
__global__ __launch_bounds__(128) void ta_s_kernel(
    const float* __restrict__ x, const float* __restrict__ W3,
    const float* __restrict__ bias, float* __restrict__ S) {
  const int bn = blockIdx.x;           // b*N + n
  const int n  = bn & (NN - 1);        // N == T, bias row
  const int t4 = threadIdx.x << 2;     // 4 t's per thread
  const v4f* xb = (const v4f*)(x + (size_t)bn * (FF * TT) + t4);

  v4f acc = (v4f)0.f;
#pragma unroll 4
  for (int f = 0; f < FF; ++f) {
    const float w = W3[f];                               // uniform -> s_load
    const v4f v = __builtin_nontemporal_load(xb + f * (TT / 4));
    acc.x = fmaf(w, v.x, acc.x);
    acc.y = fmaf(w, v.y, acc.y);
    acc.z = fmaf(w, v.z, acc.z);
    acc.w = fmaf(w, v.w, acc.w);
  }
  const v4f bv = *(const v4f*)(bias + (size_t)n * TT + t4);
  v4f r;
  r.x = 1.f / (1.f + __expf(-(acc.x + bv.x)));
  r.y = 1.f / (1.f + __expf(-(acc.y + bv.y)));
  r.z = 1.f / (1.f + __expf(-(acc.z + bv.z)));
  r.w = 1.f / (1.f + __expf(-(acc.w + bv.w)));
  *(v4f*)(S + (size_t)bn * TT + t4) = r;                 // re-read by k2: RT
}

// ---------------------------------------------------------------------------
// Kernel 2: E[b] = Vs @ S[b] via V_WMMA_F32_16X16X4_F32, softmax over E's M
// dimension (axis=1) fused in LDS. One workgroup = (b, 32-col strip); its
// 8 waves cover the full M=512 extent so E never touches HBM.
// LDS S/E strip is stored column-major: B fragments are single ds_load_b64.
// ---------------------------------------------------------------------------
__global__ __launch_bounds__(256) void ta_gemm_softmax_kernel(
    const float* __restrict__ S, const float* __restrict__ Vs,
    float* __restrict__ out) {
  __shared__ float tile[32 * LSTR];    // ~64 KB: S-strip, then reused for E
  __shared__ float red[8 * 32];        // cross-wave column reductions
  __shared__ float colv[32];           // column max, then column sum

  const int b  = blockIdx.x >> 4;          // 16 strips per batch
  const int u0 = (blockIdx.x & 15) * 32;   // column strip base
  const int tid = threadIdx.x;

  // ---- stage S[b][s][u0..u0+31] into LDS, transposed to column-major ----
  const float* Sb = S + (size_t)b * TT * TT + u0;
  for (int i = tid; i < TT * 32; i += 256) {
    const int s = i >> 5, j = i & 31;          // coalesced 128B global rows
    tile[j * LSTR + s] = Sb[(size_t)s * TT + j];   // conflict-free LDS write
  }
  __syncthreads();

  const int wave  = tid >> 5;          // wave32: 8 waves
  const int lane  = tid & 31;
  const int lhalf = lane >> 4;         // 0|1 (lanes 0-15 vs 16-31)
  const int lmod  = lane & 15;
  const int m0    = wave * 64;         // this wave's 64 rows of E

  v8f acc[4][2] = {};                  // 4 M-tiles x 2 N-tiles of 16x16 f32

  // ---- K loop over s = 0..511 in steps of 4 ----
  for (int k0 = 0; k0 < TT; k0 += 4) {
    const int krow = k0 + 2 * lhalf;   // this lane's K pair
    // B fragments: {S[k][n], S[k+1][n]} contiguous in LDS -> one b64 load
    v2f bfrag[2];
#pragma unroll
    for (int nt = 0; nt < 2; ++nt)
      bfrag[nt] = *(const v2f*)&tile[(nt * 16 + lmod) * LSTR + krow];
#pragma unroll
    for (int mt = 0; mt < 4; ++mt) {
      // A fragment (global, L2-resident Vs): contiguous 8B per lane
      const float* ap = Vs + (size_t)(m0 + mt * 16 + lmod) * TT + krow;
      v2f afrag = { ap[0], ap[1] };
      acc[mt][0] = __builtin_amdgcn_wmma_f32_16x16x4_f32(
          false, afrag, false, bfrag[0], (short)0, acc[mt][0], false, false);
      acc[mt][1] = __builtin_amdgcn_wmma_f32_16x16x4_f32(
          false, afrag, false, bfrag[1], (short)0, acc[mt][1], false, false);
    }
  }
  __syncthreads();   // everyone done reading the S strip

  // ---- dump E tiles into LDS (C/D layout: VGPR r -> row r + 8*lhalf) ----
#pragma unroll
  for (int mt = 0; mt < 4; ++mt)
#pragma unroll
    for (int nt = 0; nt < 2; ++nt)
#pragma unroll
      for (int r = 0; r < 8; ++r)
        tile[(nt * 16 + lmod) * LSTR + (m0 + mt * 16 + lhalf * 8 + r)] =
            acc[mt][nt][r];
  __syncthreads();

  // ---- column softmax over t = 0..511 ----
  const int j    = tid & 31;   // column within strip
  const int part = tid >> 5;   // 8 row-partitions of 64
  const int tlo  = part * 64;
  float* col = &tile[j * LSTR];

  float lmax = -3.402823466e+38f;
  for (int t = tlo; t < tlo + 64; ++t)
    lmax = fmaxf(lmax, col[t]);
  red[part * 32 + j] = lmax;
  __syncthreads();
  if (tid < 32) {
    float m = red[j];
#pragma unroll
    for (int p = 1; p < 8; ++p) m = fmaxf(m, red[p * 32 + j]);
    colv[j] = m;
  }
  __syncthreads();
  const float cmax = colv[j];

  float lsum = 0.f;
  for (int t = tlo; t < tlo + 64; ++t) {
    const float e = __expf(col[t] - cmax);
    col[t] = e;
    lsum += e;
  }
  red[part * 32 + j] = lsum;
  __syncthreads();
  if (tid < 32) {
    float s = 0.f;
#pragma unroll
    for (int p = 0; p < 8; ++p) s += red[p * 32 + j];
    colv[j] = s;
  }
  __syncthreads();
  const float inv = 1.f / colv[j];

  float* ob = out + (size_t)b * TT * TT + u0;
  for (int t = tlo; t < tlo + 64; ++t)                 // coalesced 128B rows
    __builtin_nontemporal_store(col[t] * inv, &ob[(size_t)t * TT + j]);
}

// ---------------------------------------------------------------------------
extern "C" void kernel_launch(void* const* d_in, const int* in_sizes, int n_in,
                              void* d_out, int out_size, void* d_ws, size_t ws_size,
                              hipStream_t stream) {
  (void)in_sizes; (void)n_in; (void)out_size; (void)ws_size;
  const float* x    = (const float*)d_in[0];
  // d_in[1] = W1, d_in[2] = W2 : dead code in reference, unused
  const float* W3   = (const float*)d_in[3];
  const float* bias = (const float*)d_in[4];
  const float* Vs   = (const float*)d_in[5];
  float* out = (float*)d_out;
  float* S   = (float*)d_ws;   // B*T*T floats = 16 MB scratch

  ta_s_kernel<<<BB * NN, 128, 0, stream>>>(x, W3, bias, S);
  ta_gemm_softmax_kernel<<<BB * 16, 256, 0, stream>>>(S, Vs, out);
}